// Sensitivity_66365834657893
// MI455X (gfx1250) — compile-verified
//
#include <hip/hip_runtime.h>
#include <hip/hip_bf16.h>
#include <stdint.h>

#define C_CLASSES  100
#define TILE_ROWS  128
#define BLOCK      128
#define ROW_F4     25                 // 100 floats = 25 float4 per row
#define ROW_BYTES  400
#define CHUNKS     25                 // TILE_ROWS*ROW_BYTES / (BLOCK*16) = 25 b128 copies/thread/tile

// ---- CDNA5 async copy: global -> LDS, 16B per lane, non-temporal (streamed-once data > L2) ----
__device__ __forceinline__ void async_copy_b128(uint32_t lds_addr, const void* gaddr) {
    asm volatile("global_load_async_to_lds_b128 %0, %1, off th:TH_LOAD_NT"
                 :: "v"(lds_addr), "v"(gaddr)
                 : "memory");
}

template <int N>
__device__ __forceinline__ void wait_asynccnt() {
    asm volatile("s_wait_asynccnt %0" :: "n"(N) : "memory");
}

__global__ void __launch_bounds__(256)
zero_counters_kernel(unsigned int* g, int n) {
    int i = blockIdx.x * blockDim.x + threadIdx.x;
    if (i < n) g[i] = 0u;
}

__global__ void __launch_bounds__(BLOCK)
sensitivity_main_kernel(const float* __restrict__ y_pred,
                        const int*   __restrict__ y_true,
                        unsigned int* __restrict__ g_cnt,   // [0..99]=colsum, [100..199]=tp
                        int n_rows) {
    __shared__ float4       s_buf[2][TILE_ROWS * ROW_F4];   // 2 x 50 KB
    __shared__ unsigned int s_colsum[C_CLASSES];
    __shared__ unsigned int s_tp[C_CLASSES];

    const int t = threadIdx.x;
    if (t < C_CLASSES) { s_colsum[t] = 0u; s_tp[t] = 0u; }
    __syncthreads();

    const int n_tiles = (n_rows + TILE_ROWS - 1) / TILE_ROWS;
    const int stride  = gridDim.x;
    const uint64_t totalBytes = (uint64_t)n_rows * ROW_BYTES;

    uint32_t lds_base[2];
    lds_base[0] = (uint32_t)(uintptr_t)(&s_buf[0][0]);
    lds_base[1] = (uint32_t)(uintptr_t)(&s_buf[1][0]);

    // Uniform issue count (always 25 asyncs/wave/tile): tail addresses are clamped
    // in-bounds so waits below can use exact immediate counts.
    auto issue_tile = [&](int tl, int buf) {
        const uint64_t base = (uint64_t)tl * (TILE_ROWS * ROW_BYTES);
#pragma unroll
        for (int i = 0; i < CHUNKS; ++i) {
            const uint32_t loff = (uint32_t)(t * 16 + i * (BLOCK * 16));
            uint64_t goff = base + loff;
            if (goff + 16 > totalBytes) goff = totalBytes - 16;   // clamp, stays issued (uniform)
            async_copy_b128(lds_base[buf] + loff,
                            (const void*)((const char*)y_pred + goff));
        }
    };

    int tile = blockIdx.x;
    int cur  = 0;
    if (tile < n_tiles) issue_tile(tile, 0);

    while (tile < n_tiles) {
        const int  nxt      = tile + stride;
        const bool has_next = (nxt < n_tiles);
        if (has_next) issue_tile(nxt, cur ^ 1);

        // In-order async completion: wait for current tile's 25, leave next tile's 25 in flight.
        if (has_next) wait_asynccnt<CHUNKS>(); else wait_asynccnt<0>();
        __syncthreads();   // every wave has waited -> whole tile visible in LDS

        const int row = tile * TILE_ROWS + t;
        if (row < n_rows) {
            const float4* rp = &s_buf[cur][t * ROW_F4];
            float best = -__builtin_huge_valf();
            int   bi   = 0;
#pragma unroll
            for (int k = 0; k < ROW_F4; ++k) {
                const float4 v = rp[k];
                const int kb = 4 * k;
                if (v.x > best) { best = v.x; bi = kb;     }   // strict '>' keeps first
                if (v.y > best) { best = v.y; bi = kb + 1; }   // occurrence, matching
                if (v.z > best) { best = v.z; bi = kb + 2; }   // jnp.argmax semantics
                if (v.w > best) { best = v.w; bi = kb + 3; }
            }
            const int label = y_true[row];
            atomicAdd(&s_colsum[bi], 1u);
            if (label == bi) atomicAdd(&s_tp[bi], 1u);
        }
        __syncthreads();   // done reading buf[cur] before it is refilled
        cur ^= 1;
        tile = nxt;
    }

    if (t < C_CLASSES) {
        atomicAdd(&g_cnt[t],             s_colsum[t]);
        atomicAdd(&g_cnt[C_CLASSES + t], s_tp[t]);
    }
}

__global__ void __launch_bounds__(128)
sensitivity_final_kernel(const unsigned int* __restrict__ g_cnt,
                         float* __restrict__ out) {
    __shared__ float red[128];
    const int t = threadIdx.x;
    float v = 0.0f;
    if (t < C_CLASSES) {
        const float colsum = (float)g_cnt[t];                // TP + FP (rows predicted as t)
        const float tp     = (float)g_cnt[C_CLASSES + t];
        v = tp / (colsum + 1e-07f);
    }
    red[t] = v;
    __syncthreads();
#pragma unroll
    for (int s = 64; s > 0; s >>= 1) {
        if (t < s) red[t] += red[t + s];
        __syncthreads();
    }
    if (t == 0) out[0] = red[0] / (float)C_CLASSES;
}

extern "C" void kernel_launch(void* const* d_in, const int* in_sizes, int n_in,
                              void* d_out, int out_size, void* d_ws, size_t ws_size,
                              hipStream_t stream) {
    const float* y_pred = (const float*)d_in[0];
    const int*   y_true = (const int*)d_in[1];
    const int n = in_sizes[1];                      // number of rows
    unsigned int* g_cnt = (unsigned int*)d_ws;      // 200 u32 counters

    zero_counters_kernel<<<1, 256, 0, stream>>>(g_cnt, 2 * C_CLASSES);

    const int n_tiles = (n + TILE_ROWS - 1) / TILE_ROWS;
    int blocks = n_tiles < 2048 ? n_tiles : 2048;   // ~7.6 tiles/block at N=2M -> real double-buffer reuse
    if (blocks > 0) {
        sensitivity_main_kernel<<<blocks, BLOCK, 0, stream>>>(y_pred, y_true, g_cnt, n);
    }
    sensitivity_final_kernel<<<1, 128, 0, stream>>>(g_cnt, (float*)d_out);
}